// LastKnowledge_50276887167554
// MI455X (gfx1250) — compile-verified
//
#include <hip/hip_runtime.h>

// Problem constants (fixed by the reference): B=64, S=512, V=256, C=3.
#define S_DIM 512
#define V_DIM 256

// 12-byte frame record (class, x, y). align(4) so merged b96 loads are legal
// under DWORD alignment mode (frame addresses are 4-byte aligned, 12-byte strided).
struct __align__(4) F3 { float c, x, y; };

__global__ __launch_bounds__(256)
void lastknow_kernel(const float* __restrict__ batch,
                     float* __restrict__ out,
                     int total /* = B*V */) {
    int idx = blockIdx.x * blockDim.x + threadIdx.x;   // one thread per (b, v)
    if (idx >= total) return;

    const int v = idx & (V_DIM - 1);   // consecutive lanes -> consecutive vehicles
    const int b = idx >> 8;            // (12-byte strided, coalesces at line level)

    const long frame = (long)V_DIM * 3;                       // floats between frames
    const float* base = batch + ((long)b * S_DIM * V_DIM + v) * 3;

    float x = 0.0f, y = 0.0f;

    // Backward scan, 4 frames per iteration: 4 independent global_load_b96 in
    // flight, plus gfx1250 global_prefetch_b8 one chunk ahead to break the
    // control-dependent load chain. Expected iterations per wave: ~2-3.
    for (int s = S_DIM - 1; s >= 3; s -= 4) {
        const float* p0 = base + (long)s * frame;
        const float* p1 = p0 - frame;
        const float* p2 = p1 - frame;
        const float* p3 = p2 - frame;

        if (s >= 8) {  // prefetch next chunk's cachelines (speculative, RT)
            __builtin_prefetch(p0 - 4 * frame, 0, 1);
            __builtin_prefetch(p1 - 4 * frame, 0, 1);
            __builtin_prefetch(p2 - 4 * frame, 0, 1);
            __builtin_prefetch(p3 - 4 * frame, 0, 1);
        }

        F3 f0 = *(const F3*)p0;
        F3 f1 = *(const F3*)p1;
        F3 f2 = *(const F3*)p2;
        F3 f3 = *(const F3*)p3;

        bool hit = true;
        if      (f0.c != -1.0f) { x = f0.x; y = f0.y; }   // latest frame wins
        else if (f1.c != -1.0f) { x = f1.x; y = f1.y; }
        else if (f2.c != -1.0f) { x = f2.x; y = f2.y; }
        else if (f3.c != -1.0f) { x = f3.x; y = f3.y; }
        else hit = false;
        if (hit) break;   // per-lane exit; wave leaves loop when EXEC drains
    }

    // (1, x, y) — x,y stay 0 if the vehicle was never valid. One b96 store.
    F3 o; o.c = 1.0f; o.x = x; o.y = y;
    *(F3*)(out + (long)idx * 3) = o;
}

extern "C" void kernel_launch(void* const* d_in, const int* in_sizes, int n_in,
                              void* d_out, int out_size, void* d_ws, size_t ws_size,
                              hipStream_t stream) {
    const float* batch = (const float*)d_in[0];   // (B, S, V, 3) float32
    float* out = (float*)d_out;                   // (B, V, 3) float32

    const int total = out_size / 3;               // B * V (16384 for B=64)
    const int block = 256;                        // 8 wave32s per block
    const int grid  = (total + block - 1) / block;

    hipLaunchKernelGGL(lastknow_kernel, dim3(grid), dim3(block), 0, stream,
                       batch, out, total);
}